// CovPooling_9852654977356
// MI455X (gfx1250) — compile-verified
//
#include <hip/hip_runtime.h>

typedef __attribute__((ext_vector_type(2))) float v2f;
typedef __attribute__((ext_vector_type(8))) float v8f;

#define D      64
#define NPG    512
#define KC     32
#define LSTR   72                 // padded LDS row stride (floats): conflict-free fragment reads
#define NCHUNK (NPG / KC)         // 16
#define OUTG   (D * (D + 1) / 2)  // 2080

// Async global -> LDS copy, 16 bytes per lane, tracked by ASYNCcnt (CDNA5).
// ldsoff = wave-relative LDS byte offset (low 32 bits of generic shared ptr).
__device__ __forceinline__ void async_copy_b128(unsigned int ldsoff, const float* gptr) {
    asm volatile("global_load_async_to_lds_b128 %0, %1, off"
                 :: "v"(ldsoff), "v"((unsigned long long)(uintptr_t)gptr)
                 : "memory");
}

__device__ __forceinline__ void wait_asynccnt0() {
    asm volatile("s_wait_asynccnt 0x0" ::: "memory");
}

__global__ __launch_bounds__(128, 4) void cov_wmma_kernel(
    const float* __restrict__ X, float* __restrict__ out, int nblocks)
{
    __shared__ __align__(16) float Xs[2][KC * LSTR];
    __shared__ float ssum[2][D];

    const int tid  = threadIdx.x;
    const int lane = tid & 31;
    const int w    = tid >> 5;        // wave id 0..3 -> output row tile
    const int g    = blockIdx.x;
    const int half = lane >> 4;       // 0/1: K-pair select (A/B layout), M+8 (C layout)
    const int sub  = lane & 15;
    const int fm   = tid & 63;        // feature index for column sums
    const int fh   = tid >> 6;        // row-half for column sums

    const float* __restrict__ Xg = X + (size_t)g * (NPG * D);
    const unsigned int lds_base = (unsigned int)(uintptr_t)(&Xs[0][0]);

    v8f acc0 = {0}, acc1 = {0}, acc2 = {0}, acc3 = {0};
    float psum = 0.f;

    // preload chunk 0 into buffer 0 via async b128 (no VGPR round-trip)
    #pragma unroll
    for (int q = 0; q < 4; ++q) {
        int item = tid + q * 128;
        int row = item >> 4, c4 = item & 15;
        async_copy_b128(lds_base + (unsigned int)((row * LSTR + c4 * 4) * 4),
                        Xg + row * D + c4 * 4);
    }
    wait_asynccnt0();
    __syncthreads();

    for (int c = 0; c < NCHUNK; ++c) {
        const int cur = c & 1;
        const bool more = (c + 1 < NCHUNK);
        if (more) {
            // kick off async fill of the other buffer; overlaps with WMMA below
            const float* src = Xg + (size_t)(c + 1) * KC * D;
            const unsigned int bbase =
                lds_base + (unsigned int)((cur ^ 1) * (KC * LSTR) * 4);
            #pragma unroll
            for (int q = 0; q < 4; ++q) {
                int item = tid + q * 128;
                int row = item >> 4, c4 = item & 15;
                async_copy_b128(bbase + (unsigned int)((row * LSTR + c4 * 4) * 4),
                                src + row * D + c4 * 4);
            }
        }

        const float* Xc = Xs[cur];
        // G += X_chunk^T * X_chunk  via V_WMMA_F32_16X16X4_F32
        #pragma unroll
        for (int kk = 0; kk < KC; kk += 4) {
            const float* r0 = Xc + (kk + 2 * half) * LSTR;  // K = 2*half
            const float* r1 = r0 + LSTR;                    // K = 2*half + 1
            v2f a, b0, b1, b2, b3;
            a.x  = r0[w * 16 + sub];  a.y  = r1[w * 16 + sub];  // A[m][k] = X[k][16w+m]
            b0.x = r0[sub];           b0.y = r1[sub];           // B[k][n] = X[k][16t+n]
            b1.x = r0[16 + sub];      b1.y = r1[16 + sub];
            b2.x = r0[32 + sub];      b2.y = r1[32 + sub];
            b3.x = r0[48 + sub];      b3.y = r1[48 + sub];
            acc0 = __builtin_amdgcn_wmma_f32_16x16x4_f32(false, a, false, b0, (short)0, acc0, false, false);
            acc1 = __builtin_amdgcn_wmma_f32_16x16x4_f32(false, a, false, b1, (short)0, acc1, false, false);
            acc2 = __builtin_amdgcn_wmma_f32_16x16x4_f32(false, a, false, b2, (short)0, acc2, false, false);
            acc3 = __builtin_amdgcn_wmma_f32_16x16x4_f32(false, a, false, b3, (short)0, acc3, false, false);
        }

        // column sums for the mean (each thread: one feature, half the rows)
        {
            const float* colp = Xc + (fh * 16) * LSTR + fm;
            #pragma unroll
            for (int r = 0; r < 16; ++r) psum += colp[r * LSTR];
        }

        if (more) wait_asynccnt0();   // our async writes into buf^1 have landed
        __syncthreads();              // everyone's writes visible before next read
    }

    ssum[fh][fm] = psum;
    __syncthreads();

    const float inv_n = 1.0f / (float)NPG;
    float* outg = out + (size_t)g * OUTG;

    // C/D layout: VGPR r, lanes 0-15 -> M=r, lanes 16-31 -> M=r+8; N = lane&15
    #pragma unroll
    for (int r = 0; r < 8; ++r) {
        const int i = w * 16 + 8 * half + r;
        const float mu_i = (ssum[0][i] + ssum[1][i]) * inv_n;
        #pragma unroll
        for (int t = 0; t < 4; ++t) {
            const int j = t * 16 + sub;
            if (i <= j) {
                const float mu_j = (ssum[0][j] + ssum[1][j]) * inv_n;
                const float gv = (t == 0 ? acc0[r] : t == 1 ? acc1[r] : t == 2 ? acc2[r] : acc3[r]);
                const float v = gv * inv_n - mu_i * mu_j;
                const int idx = i * D - (i * (i - 1)) / 2 + (j - i);
                outg[idx] = v;
            }
        }
    }
    (void)nblocks;
}

extern "C" void kernel_launch(void* const* d_in, const int* in_sizes, int n_in,
                              void* d_out, int out_size, void* d_ws, size_t ws_size,
                              hipStream_t stream) {
    const float* X = (const float*)d_in[0];
    float* out = (float*)d_out;
    const int B = in_sizes[0] / (NPG * D);   // 4096 for the reference setup
    cov_wmma_kernel<<<B, 128, 0, stream>>>(X, out, B);
}